// FeatureGrid2DEmbedding_9345848836388
// MI455X (gfx1250) — compile-verified
//
#include <hip/hip_runtime.h>
#include <math.h>
#include <stdint.h>

typedef float v4f __attribute__((ext_vector_type(4)));
typedef int   i32x4 __attribute__((ext_vector_type(4)));
typedef int   i32x8 __attribute__((ext_vector_type(8)));

constexpr int B_ = 8;
constexpr int C_ = 128;
constexpr int H_ = 128;
constexpr int W_ = 128;
constexpr int P_ = 32768;
constexpr int HW_ = H_ * W_;

// ---------------------------------------------------------------------------
// Pass 1: transpose feat_grid [B, C, HW] -> [B, HW, C] using the Tensor Data
// Mover. Each block DMAs a 32(chan) x 256(spatial) f32 tile into LDS with
// tensor_load_to_lds; the D# pad feature (1 DWORD pad every 256 DWORDs) gives
// LDS rows a stride of 257 floats -> conflict-free column reads. Waves then
// write the transposed tile to global with coalesced 128B stores.
// ---------------------------------------------------------------------------
constexpr int TS_ = 256;               // spatial tile (floats, contiguous dim0)
constexpr int TC_ = 32;                // channel tile (dim1, stride HW)
constexpr int LROW_ = TS_ + 1;         // LDS row stride after TDM padding

__global__ __launch_bounds__(256) void fg_transpose_tdm_kernel(
    const float* __restrict__ fg, float* __restrict__ fgT) {
  __shared__ float tile[TC_ * LROW_];  // 32 * 257 * 4B = 32,896 B

  const int s0 = blockIdx.x * TS_;     // spatial tile origin (HW dim)
  const int c0 = blockIdx.y * TC_;     // channel tile origin
  const int b  = blockIdx.z;

  if ((threadIdx.x >> 5) == 0) {       // one wave issues the DMA per block
    const uint64_t gaddr =
        (uint64_t)(uintptr_t)(fg + ((size_t)(b * C_ + c0) * HW_ + s0));
    const uint32_t laddr = (uint32_t)(uintptr_t)(&tile[0]);  // addr[31:0] = LDS offset

    // ---- D# group 0 (128b): count=1 | lds_addr | global_addr | type=2 ----
    i32x4 g0;
    g0.x = 1;                                        // count=1 (valid), user D#
    g0.y = (int)laddr;                               // lds_addr [63:32]
    g0.z = (int)(uint32_t)(gaddr & 0xffffffffu);     // global_addr[31:0]
    g0.w = (int)((uint32_t)((gaddr >> 32) & 0x1ffffffu) | (2u << 30)); // ga[56:32] | type=2

    // ---- D# group 1 (256b) ----
    // d0: data_size=2(4B)<<16 | pad_enable<<20 | pad_interval=7(256 dw)<<22 | pad_amount=0(1 dw)
    i32x8 g1;
    g1.s0 = (2 << 16) | (1 << 20) | (7 << 22);
    g1.s1 = (int)((uint32_t)(HW_ & 0xffff) << 16);   // tensor_dim0[15:0] @bit48
    g1.s2 = (int)(((uint32_t)HW_ >> 16) | ((uint32_t)(C_ & 0xffff) << 16)); // dim0 hi | tensor_dim1 lo
    g1.s3 = (int)(((uint32_t)C_ >> 16) | ((uint32_t)TS_ << 16));            // dim1 hi | tile_dim0
    g1.s4 = TC_;                                     // tile_dim1 | tile_dim2=0
    g1.s5 = HW_;                                     // tensor_dim0_stride[31:0]
    g1.s6 = 0;                                       // stride hi | tensor_dim1_stride lo
    g1.s7 = 0;

    // 2-group form: VADDR2/3 = NULL -> tensor up to 2D (what we need).
    asm volatile("tensor_load_to_lds %0, %1"
                 :
                 : "s"(g0), "s"(g1)
                 : "memory");
    __builtin_amdgcn_s_wait_tensorcnt(0);
  }
  __syncthreads();

  // Write transposed tile: lane = channel, waves sweep spatial positions.
  const int lane = threadIdx.x & 31;
  const int wv   = threadIdx.x >> 5;   // 0..7
  float* dst = fgT + ((size_t)b * HW_ + s0) * C_ + c0;
#pragma unroll
  for (int i = 0; i < TS_ / 8; ++i) {
    const int s = wv + 8 * i;
    dst[(size_t)s * C_ + lane] = tile[lane * LROW_ + s];  // stride-257 LDS: conflict-free
  }
}

// ---------------------------------------------------------------------------
// Per-point corner/weight math (matches reference semantics exactly).
// ---------------------------------------------------------------------------
struct PointW {
  int idx[4];
  float w[4];
  bool inb;
};

__device__ __forceinline__ PointW point_weights(float x, float y) {
  PointW r;
  r.inb = (x >= 0.0f) && (y >= 0.0f) &&
          (x <= (float)(W_ - 1)) && (y <= (float)(H_ - 1));
  const int x0 = (int)floorf(x), x1 = (int)ceilf(x);
  const int y0 = (int)floorf(y), y1 = (int)ceilf(y);
  // corner order: (fx,fy), (fx,cy), (cx,fy), (cx,cy)
  const int cx[4] = {x0, x0, x1, x1};
  const int cy[4] = {y0, y1, y0, y1};
  float sum = 0.0f;
#pragma unroll
  for (int k = 0; k < 4; ++k) {
    const float dx = x - (float)cx[k];
    const float dy = y - (float)cy[k];
    const float wk = 1.0f / (sqrtf(dx * dx + dy * dy) + 1e-10f);
    r.w[k] = wk;
    sum += wk;
    r.idx[k] = cy[k] * W_ + cx[k];
  }
  const float inv = 1.0f / sum;
#pragma unroll
  for (int k = 0; k < 4; ++k) r.w[k] *= inv;
  return r;
}

// ---------------------------------------------------------------------------
// Pass 2 (fast path): gather from transposed grid [B, HW, C].
// One wave32 per point; each lane owns 4 channels (one float4).
// Per corner: 32 lanes x 16B = one contiguous 512B row (global_load_b128),
// served from the L2-resident transposed grid. NT output stores keep the
// streamed 134MB result from evicting the grid.
// ---------------------------------------------------------------------------
__global__ __launch_bounds__(256) void gather_t_kernel(
    const float* __restrict__ tk, const float* __restrict__ fgT,
    float* __restrict__ out) {
  const int wave = threadIdx.x >> 5;
  const int lane = threadIdx.x & 31;
  const long pid = (long)blockIdx.x * 8 + wave;  // 0 .. B*P-1
  const int b = (int)(pid >> 15);                // P = 2^15

  const float x = tk[pid * 2 + 0];
  const float y = tk[pid * 2 + 1];

  v4f acc = {0.0f, 0.0f, 0.0f, 0.0f};
  const PointW pw = point_weights(x, y);
  if (pw.inb) {
    const v4f* base = (const v4f*)(fgT + (size_t)b * HW_ * C_);
#pragma unroll
    for (int k = 0; k < 4; ++k) {
      const v4f v = base[(size_t)pw.idx[k] * (C_ / 4) + lane];
      acc += pw.w[k] * v;
    }
  }
  __builtin_nontemporal_store(acc, (v4f*)out + pid * (C_ / 4) + lane);
}

// ---------------------------------------------------------------------------
// Fallback (no workspace): gather directly from [B, C, H, W] with strided
// per-channel loads. Correct but uncoalesced on the grid side.
// ---------------------------------------------------------------------------
__global__ __launch_bounds__(256) void gather_d_kernel(
    const float* __restrict__ tk, const float* __restrict__ fg,
    float* __restrict__ out) {
  const int wave = threadIdx.x >> 5;
  const int lane = threadIdx.x & 31;
  const long pid = (long)blockIdx.x * 8 + wave;
  const int b = (int)(pid >> 15);

  const float x = tk[pid * 2 + 0];
  const float y = tk[pid * 2 + 1];

  float acc[4] = {0.0f, 0.0f, 0.0f, 0.0f};
  const PointW pw = point_weights(x, y);
  if (pw.inb) {
    const float* base = fg + (size_t)b * C_ * HW_;
#pragma unroll
    for (int k = 0; k < 4; ++k) {
      const float* p0 = base + pw.idx[k];
#pragma unroll
      for (int j = 0; j < 4; ++j) {
        acc[j] += pw.w[k] * p0[(size_t)(lane + 32 * j) * HW_];
      }
    }
  }
#pragma unroll
  for (int j = 0; j < 4; ++j) {
    __builtin_nontemporal_store(acc[j], out + pid * C_ + lane + 32 * j);
  }
}

// ---------------------------------------------------------------------------
extern "C" void kernel_launch(void* const* d_in, const int* in_sizes, int n_in,
                              void* d_out, int out_size, void* d_ws,
                              size_t ws_size, hipStream_t stream) {
  const float* tk = (const float*)d_in[0];   // [B, P, 2] f32
  const float* fg = (const float*)d_in[1];   // [B, C, H, W] f32
  float* out = (float*)d_out;                // [B, P, C] f32

  const size_t needed = (size_t)B_ * HW_ * C_ * sizeof(float);
  const int npoints = B_ * P_;
  const int gather_blocks = npoints / 8;     // 8 waves (points) per 256-thr block

  if (ws_size >= needed) {
    float* fgT = (float*)d_ws;
    dim3 tg(HW_ / TS_, C_ / TC_, B_);        // 64 x 4 x 8 = 2048 blocks
    fg_transpose_tdm_kernel<<<tg, 256, 0, stream>>>(fg, fgT);
    gather_t_kernel<<<gather_blocks, 256, 0, stream>>>(tk, fgT, out);
  } else {
    gather_d_kernel<<<gather_blocks, 256, 0, stream>>>(tk, fg, out);
  }
}